// CRNO2d_52390011076819
// MI455X (gfx1250) — compile-verified
//
#include <hip/hip_runtime.h>
#include <hip/hip_bf16.h>
#include <math.h>

// ---------------------------------------------------------------------------
// FNO-2D for MI455X (gfx1250, wave32). Heavy math on V_WMMA_F32_16X16X4_F32
// (fp32 matrix pipe) via truncated-DFT GEMMs + implicit-im2col conv GEMMs.
// GEMM B panels staged into LDS with the Tensor Data Mover
// (tensor_load_to_lds + s_wait_tensorcnt). Each GEMM wave computes a 16x32
// output strip (2 N-tiles), reusing every global A fragment across 2 WMMAs.
// ---------------------------------------------------------------------------

typedef float v2f __attribute__((ext_vector_type(2)));
typedef float v8f __attribute__((ext_vector_type(8)));
typedef unsigned int u32x4 __attribute__((ext_vector_type(4)));
typedef int i32x4 __attribute__((ext_vector_type(4)));
typedef int i32x8 __attribute__((ext_vector_type(8)));

#define PI_F 3.14159265358979323846f

__device__ __forceinline__ float gelu_f(float x) {
    return 0.5f * x * (1.0f + erff(x * 0.7071067811865476f));
}

// ---------------------------------------------------------------------------
// Tensor Data Mover: 2-D tile load (4-byte elements) global -> LDS.
// Tile: `rows` rows of `tile_w` elements, row stride `stride_elems` in the
// source; packed row-major contiguously at lds_dst. D# layout per CDNA5 ISA
// ch.8 (group0: count/lds_addr/global_addr/type, group1: data_size/dims).
// ---------------------------------------------------------------------------
#if __has_builtin(__builtin_amdgcn_tensor_load_to_lds)
#define HAVE_TDM 1
__device__ __forceinline__ void tdm_load_tile_2d(
    void* lds_dst, const void* gsrc,
    unsigned tile_w, unsigned rows, unsigned long long stride_elems)
{
    const unsigned lds_off = (unsigned)(unsigned long long)(uintptr_t)lds_dst;
    const unsigned long long ga = (unsigned long long)(uintptr_t)gsrc;
    const unsigned long long str = stride_elems;
    const unsigned td0 = (unsigned)((str > tile_w) ? str : tile_w);
    const unsigned td1 = rows;

    u32x4 g0;
    g0[0] = 1u;                                            // count = 1 (valid)
    g0[1] = lds_off;                                       // lds_addr (bytes)
    g0[2] = (unsigned)(ga & 0xFFFFFFFFu);                  // global_addr lo
    g0[3] = (unsigned)((ga >> 32) & 0x1FFFFFFu)            // global_addr hi
            | 0x80000000u;                                 // type = 2 (image)
    i32x8 g1;
    g1[0] = (int)(2u << 16);                               // data_size = 4B
    g1[1] = (int)((td0 & 0xFFFFu) << 16);                  // tensor_dim0 lo16
    g1[2] = (int)(((td0 >> 16) & 0xFFFFu) | ((td1 & 0xFFFFu) << 16));
    g1[3] = (int)(((td1 >> 16) & 0xFFFFu) | ((tile_w & 0xFFFFu) << 16));
    g1[4] = (int)(rows & 0xFFFFu);                         // tile_dim1; dim2=0
    g1[5] = (int)(str & 0xFFFFFFFFull);                    // dim0_stride lo
    g1[6] = (int)((str >> 32) & 0xFFFFull);                // dim0_stride hi
    g1[7] = 0;
    i32x4 z4 = {0, 0, 0, 0};
#if defined(__clang_major__) && (__clang_major__ >= 23)
    i32x8 z8 = {0, 0, 0, 0, 0, 0, 0, 0};
    __builtin_amdgcn_tensor_load_to_lds(g0, g1, z4, z4, z8, 0);
#else
    __builtin_amdgcn_tensor_load_to_lds(g0, g1, z4, z4, 0);
#endif
}
#else
#define HAVE_TDM 0
#endif

// ---------------------------------------------------------------------------
// Workspace layout (units: floats). Total ~26.1M floats ~= 104 MB.
// ---------------------------------------------------------------------------
static const size_t OFF_A    = 0;            // 4,194,304  activation h
static const size_t OFF_B    = 4194304;      // 4,194,304  tmp1
static const size_t OFF_C    = 8388608;      // 4,194,304  tmp2
static const size_t OFF_S1   = 12582912;     // 1,048,576  stage1 (row rFFT)
static const size_t OFF_S2   = 13631488;     //   262,144  stage2 (col FFT)
static const size_t OFF_S3   = 13893632;     //   262,144  stage3 (mode mix)
static const size_t OFF_S4   = 14155776;     // 1,310,720  stage4a (col iFFT)
static const size_t OFF_POUT = 15466496;     // 10,485,760 proj spectral out
static const size_t OFF_DFT  = 25952256;     //    49,152  DFT matrices
static const size_t OFF_STAT = 26001408;     //        64  BN mean/var

// ---------------------------------------------------------------------------
// fp32 WMMA GEMM: C(MxN)=A(MxK)*B(KxN), row-major, M%64==0, N%32==0, K%8==0,
// K<=256. Block = 128 thr = 4 waves; each wave one 16x32 C strip (2 N-tiles
// sharing each A fragment). 32-column B panel staged in LDS by the TDM.
// grid = (N/32, M/64).
// ---------------------------------------------------------------------------
__global__ __launch_bounds__(128) void wmma_gemm_f32(
    const float* __restrict__ A, int lda,
    const float* __restrict__ B, int ldb,
    float* __restrict__ C, int ldc, int K)
{
    __shared__ float ldsB[32 * 256];           // B panel: [k][n], k<=256, n=32
    const int lane = threadIdx.x & 31;
    const int wave = threadIdx.x >> 5;
    const int half = lane >> 4;
    const int r    = lane & 15;
    const int mb = (blockIdx.y * 4 + wave) * 16;
    const int nb = blockIdx.x * 32;

#if HAVE_TDM
    if (wave == 0) {   // one wave drives the DMA; EXEC ignored by TDM
        tdm_load_tile_2d(ldsB, B + nb, 32u, (unsigned)K,
                         (unsigned long long)ldb);
        __builtin_amdgcn_s_wait_tensorcnt(0);
    }
#else
    for (int i = threadIdx.x; i < K * 32; i += 128) {
        const int k = i >> 5, n = i & 31;
        ldsB[i] = B[(size_t)k * ldb + nb + n];
    }
#endif
    __syncthreads();

    const float* ap = A + (size_t)(mb + r) * lda;
    v8f acc0 = {}, acc1 = {};
    for (int k0 = 0; k0 < K; k0 += 8) {
        __builtin_prefetch(ap + k0 + 8, 0, 3);
#pragma unroll
        for (int s = 0; s < 2; ++s) {
            const int ka = k0 + 4 * s + 2 * half;
            v2f a, b0, b1;
            a.x  = ap[ka + 0];
            a.y  = ap[ka + 1];
            b0.x = ldsB[(ka + 0) * 32 + r];
            b0.y = ldsB[(ka + 1) * 32 + r];
            b1.x = ldsB[(ka + 0) * 32 + 16 + r];
            b1.y = ldsB[(ka + 1) * 32 + 16 + r];
            acc0 = __builtin_amdgcn_wmma_f32_16x16x4_f32(
                false, a, false, b0, (short)0, acc0, false, false);
            acc1 = __builtin_amdgcn_wmma_f32_16x16x4_f32(
                false, a, false, b1, (short)0, acc1, false, false);
        }
    }
#pragma unroll
    for (int j = 0; j < 8; ++j) {
        float* cr = C + (size_t)(mb + j + 8 * half) * ldc + nb + r;
        cr[0]  = acc0[j];
        cr[16] = acc1[j];
    }
}

// ---------------------------------------------------------------------------
// 3x3 conv as implicit-im2col WMMA GEMM. Per image: Out(Co x HW) =
// Wgt(Co x Ci*9) * Patches(Ci*9 x HW). 16-row weight panel staged in LDS by
// the TDM. grid = (HW/64, Co/16, Batch). K = Ci*9 <= 288, K%8==0.
// ---------------------------------------------------------------------------
__global__ __launch_bounds__(128) void wmma_conv3x3(
    const float* __restrict__ in, const float* __restrict__ wgt,
    const float* __restrict__ bias, float* __restrict__ out,
    int Ci, int Hd, int Wd, int do_gelu)
{
    __shared__ float ldsA[16 * 288];            // weight panel: [m][k]
    const int lane = threadIdx.x & 31;
    const int wave = threadIdx.x >> 5;
    const int half = lane >> 4;
    const int r    = lane & 15;
    const int bimg = blockIdx.z;
    const int cb   = blockIdx.y * 16;
    const int pb   = (blockIdx.x * 4 + wave) * 16;
    const int HW   = Hd * Wd;
    const int K    = Ci * 9;
    const int Co   = gridDim.y * 16;
    const float* inb = in + (size_t)bimg * Ci * HW;

#if HAVE_TDM
    if (wave == 0) {
        tdm_load_tile_2d(ldsA, wgt + (size_t)cb * K, (unsigned)K, 16u,
                         (unsigned long long)K);
        __builtin_amdgcn_s_wait_tensorcnt(0);
    }
#else
    for (int i = threadIdx.x; i < 16 * K; i += 128) {
        const int m = i / K, k = i - m * K;
        ldsA[i] = wgt[(size_t)(cb + m) * K + k];
    }
#endif
    __syncthreads();

    const int p  = pb + r;
    const int py = p / Wd, px = p - py * Wd;

    v8f acc = {};
    for (int k0 = 0; k0 < K; k0 += 8) {
#pragma unroll
        for (int s = 0; s < 2; ++s) {
            const int ka = k0 + 4 * s + 2 * half;
            v2f a, b;
            a.x = ldsA[r * K + ka + 0];
            a.y = ldsA[r * K + ka + 1];
#pragma unroll
            for (int j = 0; j < 2; ++j) {  // im2col gather for (k, pixel)
                const int k  = ka + j;
                const int ci = k / 9, rem = k - ci * 9;
                const int dy = rem / 3 - 1, dx = rem - (rem / 3) * 3 - 1;
                const int iy = py + dy, ix = px + dx;
                float v = 0.0f;
                if (iy >= 0 && iy < Hd && ix >= 0 && ix < Wd)
                    v = inb[(size_t)ci * HW + iy * Wd + ix];
                if (j == 0) b.x = v; else b.y = v;
            }
            acc = __builtin_amdgcn_wmma_f32_16x16x4_f32(
                false, a, false, b, (short)0, acc, false, false);
        }
    }
#pragma unroll
    for (int j = 0; j < 8; ++j) {
        const int co = cb + j + 8 * half;
        float v = acc[j] + bias[co];
        if (do_gelu) v = gelu_f(v);
        out[((size_t)bimg * Co + co) * HW + p] = v;
    }
}

// ---------------------------------------------------------------------------
// Input lift: h[:,0]=x, h[:,1]=linspace over H, h[:,2]=linspace over W.
// ---------------------------------------------------------------------------
__global__ void build_input_kernel(const float* __restrict__ x,
                                   float* __restrict__ out, int B, int Hd, int Wd)
{
    const int HW = Hd * Wd;
    int idx = blockIdx.x * blockDim.x + threadIdx.x;
    if (idx >= B * HW) return;
    const int b = idx / HW, p = idx - b * HW;
    const int y = p / Wd, xx = p - y * Wd;
    out[((size_t)b * 3 + 0) * HW + p] = x[idx];
    out[((size_t)b * 3 + 1) * HW + p] = (float)y  / (float)(Hd - 1);
    out[((size_t)b * 3 + 2) * HW + p] = (float)xx / (float)(Wd - 1);
}

// ---------------------------------------------------------------------------
// DFT matrix builders (ortho norm folded in).
// ---------------------------------------------------------------------------
__global__ void init_F_kernel(float* __restrict__ F, int N) {
    int t = blockIdx.x * blockDim.x + threadIdx.x;
    if (t >= N * 16) return;
    const int w = t / 16, ky = t - w * 16;
    const float s = rsqrtf((float)N);
    const float ang = 2.0f * PI_F * (float)w * (float)ky / (float)N;
    F[w * 32 + 2 * ky + 0] =  cosf(ang) * s;
    F[w * 32 + 2 * ky + 1] = -sinf(ang) * s;
}
__global__ void init_G_kernel(float* __restrict__ G, int N) {
    int t = blockIdx.x * blockDim.x + threadIdx.x;
    if (t >= 31 * N) return;
    const int kxi = t / N, h = t - kxi * N;
    const int kx = kxi - 15;
    const float s = rsqrtf((float)N);
    const float ang = 2.0f * PI_F * (float)h * (float)kx / (float)N;
    G[2 * t + 0] =  cosf(ang) * s;
    G[2 * t + 1] = -sinf(ang) * s;
}
__global__ void init_H_kernel(float* __restrict__ Hm, int M) {
    int t = blockIdx.x * blockDim.x + threadIdx.x;
    if (t >= M * 31) return;
    const int u = t / 31, kxi = t - u * 31;
    const int kx = kxi - 15;
    const float s = rsqrtf((float)M);
    const float ang = 2.0f * PI_F * (float)u * (float)kx / (float)M;
    Hm[2 * t + 0] = cosf(ang) * s;
    Hm[2 * t + 1] = sinf(ang) * s;
}
__global__ void init_E_kernel(float* __restrict__ E, int M, float scale) {
    int t = blockIdx.x * blockDim.x + threadIdx.x;
    if (t >= 16 * M) return;
    const int ky = t / M, v = t - ky * M;
    const float c = (ky == 0) ? 1.0f : 2.0f;
    const float ang = 2.0f * PI_F * (float)ky * (float)v / (float)M;
    E[(size_t)(2 * ky + 0) * M + v] =  c * cosf(ang) * scale;
    E[(size_t)(2 * ky + 1) * M + v] = -c * sinf(ang) * scale;
}

// ---------------------------------------------------------------------------
// Stage2: column FFT (K=N) restricted to 31 output freqs.
// ---------------------------------------------------------------------------
__global__ void stage2_kernel(const float* __restrict__ S1,
                              const float* __restrict__ G,
                              float* __restrict__ S2, int N, int nimg)
{
    int t = blockIdx.x * blockDim.x + threadIdx.x;
    if (t >= nimg * 31 * 16) return;
    const int ky = t & 15;
    int q = t >> 4;
    const int kxi = q % 31;
    const int img = q / 31;
    float re = 0.0f, im = 0.0f;
    const float* gp = G + (size_t)kxi * N * 2;
    const float* sp = S1 + (size_t)img * N * 32 + 2 * ky;
    for (int h = 0; h < N; ++h) {
        const float gr = gp[2 * h], gi = gp[2 * h + 1];
        const float xr = sp[(size_t)h * 32], xi = sp[(size_t)h * 32 + 1];
        re += gr * xr - gi * xi;
        im += gr * xi + gi * xr;
    }
    S2[2 * t + 0] = re;
    S2[2 * t + 1] = im;
}

// ---------------------------------------------------------------------------
// Stage3: per-mode channel mixing; complex weight assembled on the fly.
// ---------------------------------------------------------------------------
__global__ void stage3_kernel(const float* __restrict__ S2,
                              const float* __restrict__ y0r,
                              const float* __restrict__ y0i,
                              const float* __restrict__ w00,
                              const float* __restrict__ yxr,
                              const float* __restrict__ yxi,
                              float* __restrict__ S3, int B, int Ci, int Co)
{
    int t = blockIdx.x * blockDim.x + threadIdx.x;
    if (t >= B * Co * 31 * 16) return;
    const int ky = t & 15;
    int q = t >> 4;
    const int xm = q % 31; q /= 31;
    const int o = q % Co;
    const int b = q / Co;

    float re = 0.0f, im = 0.0f;
    for (int i = 0; i < Ci; ++i) {
        float wr, wi;
        if (ky >= 1) {
            const size_t idx = (((size_t)i * Co + o) * 31 + xm) * 15 + (ky - 1);
            wr = yxr[idx]; wi = yxi[idx];
        } else if (xm < 15) {
            const size_t idx = ((size_t)i * Co + o) * 15 + xm;
            wr = y0r[idx]; wi = y0i[idx];
        } else if (xm == 15) {
            wr = w00[(size_t)i * Co + o]; wi = 0.0f;
        } else {
            const size_t idx = ((size_t)i * Co + o) * 15 + (30 - xm);
            wr = y0r[idx]; wi = -y0i[idx];
        }
        const size_t s = (((size_t)(b * Ci + i) * 31 + xm) * 16 + ky) * 2;
        const float xr = S2[s], xi = S2[s + 1];
        re += xr * wr - xi * wi;
        im += xr * wi + xi * wr;
    }
    const size_t d = (((size_t)(b * Co + o) * 31 + xm) * 16 + ky) * 2;
    S3[d] = re; S3[d + 1] = im;
}

// ---------------------------------------------------------------------------
// Stage4a: inverse column FFT (K=31); rows packed [re,im] per ky for the
// stage4b real GEMM (K=32).
// ---------------------------------------------------------------------------
__global__ void stage4a_kernel(const float* __restrict__ S3,
                               const float* __restrict__ Hm,
                               float* __restrict__ S4, int M, int nimg)
{
    int t = blockIdx.x * blockDim.x + threadIdx.x;
    if (t >= nimg * M * 16) return;
    const int ky = t & 15;
    int q = t >> 4;
    const int u = q % M;
    const int img = q / M;
    float re = 0.0f, im = 0.0f;
    const float* hp = Hm + (size_t)u * 31 * 2;
    const float* sp = S3 + ((size_t)img * 31 * 16 + ky) * 2;
    for (int kxi = 0; kxi < 31; ++kxi) {
        const float hr = hp[2 * kxi], hi = hp[2 * kxi + 1];
        const float yr = sp[(size_t)kxi * 32], yi = sp[(size_t)kxi * 32 + 1];
        re += hr * yr - hi * yi;
        im += hr * yi + hi * yr;
    }
    S4[((size_t)img * M + u) * 32 + 2 * ky + 0] = re;
    S4[((size_t)img * M + u) * 32 + 2 * ky + 1] = im;
}

// ---------------------------------------------------------------------------
// BatchNorm (population stats over B,H,W) + optional GELU.
// ---------------------------------------------------------------------------
__global__ void bn_stats_kernel(const float* __restrict__ x,
                                float* __restrict__ stat, int C, int HW, int B)
{
    __shared__ float s1[256], s2[256];
    const int c = blockIdx.x, t = threadIdx.x;
    const int n = B * HW;
    float a1 = 0.0f, a2 = 0.0f;
    for (int j = t; j < n; j += 256) {
        const int b = j / HW, p = j - b * HW;
        const float v = x[((size_t)b * C + c) * HW + p];
        a1 += v; a2 += v * v;
    }
    s1[t] = a1; s2[t] = a2;
    __syncthreads();
    for (int s = 128; s > 0; s >>= 1) {
        if (t < s) { s1[t] += s1[t + s]; s2[t] += s2[t + s]; }
        __syncthreads();
    }
    if (t == 0) {
        const float mu = s1[0] / (float)n;
        stat[2 * c]     = mu;
        stat[2 * c + 1] = s2[0] / (float)n - mu * mu;
    }
}

__global__ void bn_apply_kernel(float* __restrict__ x,
                                const float* __restrict__ stat,
                                const float* __restrict__ g,
                                const float* __restrict__ bb,
                                int C, int HW, int B, int do_gelu)
{
    int idx = blockIdx.x * blockDim.x + threadIdx.x;
    if (idx >= B * C * HW) return;
    const int c = (idx / HW) % C;
    const float mu = stat[2 * c], var = stat[2 * c + 1];
    float v = g[c] * (x[idx] - mu) * rsqrtf(var + 1e-5f) + bb[c];
    if (do_gelu) v = gelu_f(v);
    x[idx] = v;
}

__global__ void add_kernel(const float* __restrict__ a,
                           const float* __restrict__ b,
                           float* __restrict__ out, int n, int do_gelu)
{
    int idx = blockIdx.x * blockDim.x + threadIdx.x;
    if (idx >= n) return;
    float v = a[idx] + b[idx];
    if (do_gelu) v = gelu_f(v);
    out[idx] = v;
}

// ---------------------------------------------------------------------------
// Fused projection head: 1x1 (20->80) + GELU + 1x1 (80->1), per pixel.
// ---------------------------------------------------------------------------
__global__ void head_kernel(const float* __restrict__ pin,
                            const float* __restrict__ w1,
                            const float* __restrict__ b1,
                            const float* __restrict__ w2,
                            const float* __restrict__ b2,
                            float* __restrict__ out, int HW, int B)
{
    int idx = blockIdx.x * blockDim.x + threadIdx.x;
    if (idx >= B * HW) return;
    const int b = idx / HW, p = idx - b * HW;
    float v[20];
#pragma unroll
    for (int c = 0; c < 20; ++c)
        v[c] = pin[((size_t)b * 20 + c) * HW + p];
    float acc = b2[0];
    for (int j = 0; j < 80; ++j) {
        float h = b1[j];
#pragma unroll
        for (int c = 0; c < 20; ++c)
            h += w1[j * 20 + c] * v[c];
        acc += w2[j] * gelu_f(h);
    }
    out[idx] = acc;
}

// ---------------------------------------------------------------------------
// Host orchestration.
// ---------------------------------------------------------------------------
static inline dim3 grid1d(int n) { return dim3((n + 255) / 256); }

static void run_spectral(hipStream_t stream, float* ws,
                         const float* in, float* out,
                         int B, int Ci, int Co, int N, int M,
                         const float* y0r, const float* y0i, const float* w00,
                         const float* yxr, const float* yxi)
{
    float* F  = ws + OFF_DFT;            // N x 32
    float* G  = F + 8192;                // 31 x N complex
    float* Hm = G + 15872;               // M x 31 complex
    float* E  = Hm + 15872;              // 32 x M
    float* S1 = ws + OFF_S1;
    float* S2 = ws + OFF_S2;
    float* S3 = ws + OFF_S3;
    float* S4 = ws + OFF_S4;

    init_F_kernel<<<grid1d(N * 16), 256, 0, stream>>>(F, N);
    init_G_kernel<<<grid1d(31 * N), 256, 0, stream>>>(G, N);
    init_H_kernel<<<grid1d(M * 31), 256, 0, stream>>>(Hm, M);
    const float scale = ((float)M / (float)N) * rsqrtf((float)M);
    init_E_kernel<<<grid1d(16 * M), 256, 0, stream>>>(E, M, scale);

    // Stage1: row rFFT -> 16 modes.  (B*Ci*N x N) @ (N x 32)
    const int rows1 = B * Ci * N;
    wmma_gemm_f32<<<dim3(32 / 32, rows1 / 64), 128, 0, stream>>>(
        in, N, F, 32, S1, 32, N);

    stage2_kernel<<<grid1d(B * Ci * 31 * 16), 256, 0, stream>>>(
        S1, G, S2, N, B * Ci);
    stage3_kernel<<<grid1d(B * Co * 31 * 16), 256, 0, stream>>>(
        S2, y0r, y0i, w00, yxr, yxi, S3, B, Ci, Co);
    stage4a_kernel<<<grid1d(B * Co * M * 16), 256, 0, stream>>>(
        S3, Hm, S4, M, B * Co);

    // Stage4b: final real inverse row transform. (B*Co*M x 32) @ (32 x M)
    const int rows4 = B * Co * M;
    wmma_gemm_f32<<<dim3(M / 32, rows4 / 64), 128, 0, stream>>>(
        S4, 32, E, M, out, M, 32);
}

extern "C" void kernel_launch(void* const* d_in, const int* in_sizes, int n_in,
                              void* d_out, int out_size, void* d_ws, size_t ws_size,
                              hipStream_t stream)
{
    (void)in_sizes; (void)n_in; (void)out_size; (void)ws_size;
    const float* x = (const float*)d_in[0];
    float* ws = (float*)d_ws;
    float* A    = ws + OFF_A;
    float* Bb   = ws + OFF_B;
    float* Cc   = ws + OFF_C;
    float* POUT = ws + OFF_POUT;
    float* STAT = ws + OFF_STAT;

    const int Bn = 8;
    const int HW256 = 256 * 256;
    const int HW128 = 128 * 128;

    // ---- lift -------------------------------------------------------------
    build_input_kernel<<<grid1d(Bn * HW256), 256, 0, stream>>>(x, A, Bn, 256, 256);

    run_spectral(stream, ws, A, Bb, Bn, 3, 16, 256, 128,
                 (const float*)d_in[1], (const float*)d_in[2],
                 (const float*)d_in[3], (const float*)d_in[4],
                 (const float*)d_in[5]);

    bn_stats_kernel<<<16, 256, 0, stream>>>(Bb, STAT, 16, HW128, Bn);
    bn_apply_kernel<<<grid1d(Bn * 16 * HW128), 256, 0, stream>>>(
        Bb, STAT, (const float*)d_in[6], (const float*)d_in[7],
        16, HW128, Bn, 1);

    // conv3x3 16 -> 32  (Bb -> A)
    wmma_conv3x3<<<dim3(HW128 / 64, 32 / 16, Bn), 128, 0, stream>>>(
        Bb, (const float*)d_in[8], (const float*)d_in[9], A, 16, 128, 128, 0);

    // ---- 4 spectral blocks at 128^2 / 32 ch -------------------------------
    for (int i = 0; i < 4; ++i) {
        const int base = 10 + i * 11;
        run_spectral(stream, ws, A, Bb, Bn, 32, 32, 128, 128,
                     (const float*)d_in[base + 0], (const float*)d_in[base + 1],
                     (const float*)d_in[base + 2], (const float*)d_in[base + 3],
                     (const float*)d_in[base + 4]);
        // x1 = conv(gelu(conv(x1)))
        wmma_conv3x3<<<dim3(HW128 / 64, 2, Bn), 128, 0, stream>>>(
            Bb, (const float*)d_in[base + 5], (const float*)d_in[base + 6],
            Cc, 32, 128, 128, 1);
        wmma_conv3x3<<<dim3(HW128 / 64, 2, Bn), 128, 0, stream>>>(
            Cc, (const float*)d_in[base + 7], (const float*)d_in[base + 8],
            Bb, 32, 128, 128, 0);
        // x2 = conv(h)
        wmma_conv3x3<<<dim3(HW128 / 64, 2, Bn), 128, 0, stream>>>(
            A, (const float*)d_in[base + 9], (const float*)d_in[base + 10],
            Cc, 32, 128, 128, 0);
        // h = (x1 + x2), gelu except last
        add_kernel<<<grid1d(Bn * 32 * HW128), 256, 0, stream>>>(
            Bb, Cc, A, Bn * 32 * HW128, (i < 3) ? 1 : 0);
    }

    // ---- projection -------------------------------------------------------
    run_spectral(stream, ws, A, POUT, Bn, 32, 20, 128, 256,
                 (const float*)d_in[54], (const float*)d_in[55],
                 (const float*)d_in[56], (const float*)d_in[57],
                 (const float*)d_in[58]);

    bn_stats_kernel<<<20, 256, 0, stream>>>(POUT, STAT, 20, HW256, Bn);
    bn_apply_kernel<<<grid1d(Bn * 20 * HW256), 256, 0, stream>>>(
        POUT, STAT, (const float*)d_in[59], (const float*)d_in[60],
        20, HW256, Bn, 0);

    head_kernel<<<grid1d(Bn * HW256), 256, 0, stream>>>(
        POUT, (const float*)d_in[61], (const float*)d_in[62],
        (const float*)d_in[63], (const float*)d_in[64],
        (float*)d_out, HW256, Bn);
}